// IDSGNNModel_50525995270616
// MI455X (gfx1250) — compile-verified
//
#include <hip/hip_runtime.h>
#include <hip/hip_bf16.h>
#include <math.h>

typedef __attribute__((ext_vector_type(2))) float v2f;
typedef __attribute__((ext_vector_type(8))) float v8f;

// ---------------------------------------------------------------------------
// fp32 GEMM: C[M,Nd] = A[M,Kd] @ B[Kd,Nd] via V_WMMA_F32_16X16X4_F32.
// Block = 5 waves (160 thr), each wave owns one 16-row M-tile; all 5 waves
// share one 16*NT-wide N-strip of B, staged through LDS in CK-deep K-chunks.
// Compile-time Kd/Nd/NT -> all strides fold to immediate offsets.
// Requires: M % (16*5) == 0, Kd % CK == 0, Nd % (16*NT) == 0.
// ---------------------------------------------------------------------------
template <int NT, int Kd, int Nd>
__global__ __launch_bounds__(160)
void gemm_wmma_f32(const float* __restrict__ A,
                   const float* __restrict__ B,
                   float* __restrict__ C) {
    constexpr int CK = 64;            // K-chunk depth
    constexpr int NW = 16 * NT;       // N-strip width
    __shared__ float bs[CK * NW];     // 32 KB (NT=8) / 16 KB (NT=4)

    const int wave = threadIdx.x >> 5;          // 0..4
    const int lane = threadIdx.x & 31;
    const int r16  = lane & 15;                 // M index (A) / N index (B,C)
    const int koff = (lane >> 4) << 1;          // 0 | 2

    const int tileM = blockIdx.x * 5 + wave;
    const int nbase = blockIdx.y * NW;

    const float* __restrict__ Arow = A + (size_t)(tileM * 16 + r16) * Kd;

    v8f acc[NT];
#pragma unroll
    for (int t = 0; t < NT; ++t) acc[t] = (v8f){};

    for (int kc = 0; kc < Kd; kc += CK) {
        __syncthreads();   // protect bs before overwrite (no-op cost on iter 0)
        // cooperative stage of B[kc:kc+CK, nbase:nbase+NW] -> bs[CK][NW]
        for (int i = threadIdx.x; i < (CK * NW) / 4; i += 160) {
            const int row = i / (NW / 4);
            const int c4  = i % (NW / 4);
            ((float4*)bs)[i] =
                *(const float4*)&B[(size_t)(kc + row) * Nd + nbase + c4 * 4];
        }
        __syncthreads();

        for (int k0 = 0; k0 < CK; k0 += 4) {
            v2f a;
            a.x = Arow[kc + k0 + koff];
            a.y = Arow[kc + k0 + koff + 1];
            const float* __restrict__ Br0 = &bs[(k0 + koff) * NW + r16];
            const float* __restrict__ Br1 = Br0 + NW;
            v2f bf[NT];
#pragma unroll
            for (int t = 0; t < NT; ++t) {
                bf[t].x = Br0[t * 16];
                bf[t].y = Br1[t * 16];
            }
#pragma unroll
            for (int t = 0; t < NT; ++t)
                acc[t] = __builtin_amdgcn_wmma_f32_16x16x4_f32(
                    false, a, false, bf[t], (short)0, acc[t], false, false);
        }
    }

    // C/D layout: lanes 0-15 -> rows 0..7 of tile, lanes 16-31 -> rows 8..15
    const int mbase = tileM * 16 + ((lane >> 4) << 3);
#pragma unroll
    for (int t = 0; t < NT; ++t) {
        float* __restrict__ Cp = C + nbase + t * 16 + r16;
#pragma unroll
        for (int i = 0; i < 8; ++i)
            Cp[(size_t)(mbase + i) * Nd] = acc[t][i];
    }
}

// ---------------------------------------------------------------------------
// Attention coefficients: a_s[n,h] = <h[n,h,:], att_src[h,:]>, same for a_d.
// ---------------------------------------------------------------------------
__global__ void attn_coeff(const float* __restrict__ Hm,
                           const float* __restrict__ att_src,
                           const float* __restrict__ att_dst,
                           float* __restrict__ as_, float* __restrict__ ad_,
                           int Nn, int Hh, int Cc) {
    int t = blockIdx.x * blockDim.x + threadIdx.x;
    if (t >= Nn * Hh) return;
    int h = t % Hh;
    const float* hp = Hm + (size_t)t * Cc;
    const float* sp = att_src + h * Cc;
    const float* dp = att_dst + h * Cc;
    float s = 0.f, d = 0.f;
    for (int c = 0; c < Cc; ++c) {
        float v = hp[c];
        s += v * sp[c];
        d += v * dp[c];
    }
    as_[t] = s;
    ad_[t] = d;
}

__global__ void fill_f32(float* __restrict__ p, float v, int n) {
    int i = blockIdx.x * blockDim.x + threadIdx.x;
    if (i < n) p[i] = v;
}

__device__ __forceinline__ void atomicMaxF32(float* addr, float val) {
    if (val >= 0.f)
        atomicMax((int*)addr, __float_as_int(val));
    else
        atomicMin((unsigned int*)addr, __float_as_uint(val));
}

__device__ __forceinline__ void edge_endpoints(const int* __restrict__ ei,
                                               int E_, int e, int& s, int& d) {
    if (e < E_) { s = ei[e]; d = ei[E_ + e]; }
    else        { s = d = e - E_; }   // appended self-loops
}

// pass 1: per-destination max of leaky_relu(a_s[src]+a_d[dst])
__global__ void edge_max(const int* __restrict__ ei, int E_, int Etot,
                         const float* __restrict__ as_, const float* __restrict__ ad_,
                         float* __restrict__ mx, int Hh) {
    int t = blockIdx.x * blockDim.x + threadIdx.x;
    if (t >= Etot * Hh) return;
    int e = t / Hh, h = t % Hh, s, d;
    edge_endpoints(ei, E_, e, s, d);
    float v = as_[s * Hh + h] + ad_[d * Hh + h];
    v = (v > 0.f) ? v : 0.2f * v;
    atomicMaxF32(&mx[d * Hh + h], v);
}

// pass 2: alpha_raw = exp(e - max); denom += alpha_raw
__global__ void edge_expsum(const int* __restrict__ ei, int E_, int Etot,
                            const float* __restrict__ as_, const float* __restrict__ ad_,
                            const float* __restrict__ mx,
                            float* __restrict__ alpha, float* __restrict__ denom, int Hh) {
    int t = blockIdx.x * blockDim.x + threadIdx.x;
    if (t >= Etot * Hh) return;
    int e = t / Hh, h = t % Hh, s, d;
    edge_endpoints(ei, E_, e, s, d);
    float v = as_[s * Hh + h] + ad_[d * Hh + h];
    v = (v > 0.f) ? v : 0.2f * v;
    float ex = expf(v - mx[d * Hh + h]);
    alpha[t] = ex;
    atomicAdd(&denom[d * Hh + h], ex);
}

// pass 3: out[dst,h,c..c+3] += (alpha/denom) * Hm[src,h,c..c+3]
__global__ void edge_aggregate(const int* __restrict__ ei, int E_, int Etot,
                               const float* __restrict__ alpha,
                               const float* __restrict__ denom,
                               const float* __restrict__ Hm,
                               float* __restrict__ out, int Hh, int Cc) {
    int t = blockIdx.x * blockDim.x + threadIdx.x;
    const int cq = Cc >> 2;
    int total = Etot * Hh * cq;
    if (t >= total) return;
    int q  = t % cq;
    int eh = t / cq;
    int h  = eh % Hh;
    int e  = eh / Hh;
    int s, d;
    edge_endpoints(ei, E_, e, s, d);
    float a = alpha[e * Hh + h] / (denom[d * Hh + h] + 1e-16f);
    const float4 hv = *(const float4*)&Hm[((size_t)s * Hh + h) * Cc + q * 4];
    float* op = &out[((size_t)d * Hh + h) * Cc + q * 4];
    atomicAdd(op + 0, a * hv.x);
    atomicAdd(op + 1, a * hv.y);
    atomicAdd(op + 2, a * hv.z);
    atomicAdd(op + 3, a * hv.w);
}

__global__ void bias_elu(const float* __restrict__ in, const float* __restrict__ bias,
                         float* __restrict__ out, int total, int Fdim) {
    int t = blockIdx.x * blockDim.x + threadIdx.x;
    if (t >= total) return;
    float v = in[t] + bias[t % Fdim];
    out[t] = (v > 0.f) ? v : (expf(v) - 1.f);
}

__global__ void pool_add(const float* __restrict__ h, const int* __restrict__ batch,
                         float* __restrict__ pool, int Nn, int Cc) {
    int t = blockIdx.x * blockDim.x + threadIdx.x;
    if (t >= Nn * Cc) return;
    int n = t / Cc, c = t % Cc;
    atomicAdd(&pool[batch[n] * Cc + c], h[t]);
}

// MLP head + log_softmax. One block of B=64 threads, one graph per thread.
__global__ void head_kernel(const float* __restrict__ pool,
                            const float* __restrict__ w1, const float* __restrict__ b1_,
                            const float* __restrict__ w2, const float* __restrict__ b2_,
                            float* __restrict__ out) {
    const int b = threadIdx.x;   // 0..63
    float z[64];
#pragma unroll 4
    for (int j = 0; j < 64; ++j) {
        float acc = b1_[j];
        for (int i = 0; i < 64; ++i) acc += pool[b * 64 + i] * w1[i * 64 + j];
        z[j] = (acc > 0.f) ? acc : (expf(acc) - 1.f);
    }
    float logits[10];
    float mx = -INFINITY;
#pragma unroll
    for (int k = 0; k < 10; ++k) {
        float acc = b2_[k];
        for (int i = 0; i < 64; ++i) acc += z[i] * w2[i * 10 + k];
        logits[k] = acc;
        mx = fmaxf(mx, acc);
    }
    float s = 0.f;
#pragma unroll
    for (int k = 0; k < 10; ++k) s += expf(logits[k] - mx);
    float lse = mx + logf(s);
#pragma unroll
    for (int k = 0; k < 10; ++k) out[b * 10 + k] = logits[k] - lse;
}

// ---------------------------------------------------------------------------
extern "C" void kernel_launch(void* const* d_in, const int* in_sizes, int n_in,
                              void* d_out, int out_size, void* d_ws, size_t ws_size,
                              hipStream_t stream) {
    constexpr int N = 50000, E = 400000, F = 128, H = 8, C = 64, B = 64;
    const int Etot = E + N;

    const float* x     = (const float*)d_in[0];
    const int*   ei    = (const int*)  d_in[1];
    const int*   batch = (const int*)  d_in[2];
    const float* W1    = (const float*)d_in[3];
    const float* aS1   = (const float*)d_in[4];
    const float* aD1   = (const float*)d_in[5];
    const float* b1    = (const float*)d_in[6];
    const float* W2    = (const float*)d_in[7];
    const float* aS2   = (const float*)d_in[8];
    const float* aD2   = (const float*)d_in[9];
    const float* b2    = (const float*)d_in[10];
    const float* l1w   = (const float*)d_in[11];
    const float* l1b   = (const float*)d_in[12];
    const float* l2w   = (const float*)d_in[13];
    const float* l2b   = (const float*)d_in[14];
    float* out = (float*)d_out;

    // Workspace arena (fp32)
    float* ws    = (float*)d_ws;
    float* H1    = ws;                            // N*512 (feat / later h1_elu)
    float* AGG   = H1 + (size_t)N * H * C;        // N*512 (agg target; later H2/H2agg)
    float* AS    = AGG + (size_t)N * H * C;       // N*8
    float* AD    = AS + (size_t)N * H;            // N*8
    float* MX    = AD + (size_t)N * H;            // N*8
    float* DN    = MX + (size_t)N * H;            // N*8
    float* ALPHA = DN + (size_t)N * H;            // Etot*8
    float* POOL  = ALPHA + (size_t)Etot * H;      // B*64

    const int TB = 256;
    auto nb = [](int n, int tb) { return (n + tb - 1) / tb; };

    // ---------------- Layer 1 (H=8, C=64) ----------------
    // grid: (3125/5 M-groups, 512/128 N-strips), 160 threads (5 waves)
    gemm_wmma_f32<8, F, H * C><<<dim3(N / 16 / 5, (H * C) / 128), 160, 0, stream>>>(x, W1, H1);
    attn_coeff<<<nb(N * H, TB), TB, 0, stream>>>(H1, aS1, aD1, AS, AD, N, H, C);
    fill_f32<<<nb(N * H, TB), TB, 0, stream>>>(MX, -INFINITY, N * H);
    fill_f32<<<nb(N * H, TB), TB, 0, stream>>>(DN, 0.f, N * H);
    fill_f32<<<nb(N * H * C, TB), TB, 0, stream>>>(AGG, 0.f, N * H * C);
    edge_max   <<<nb(Etot * H, TB), TB, 0, stream>>>(ei, E, Etot, AS, AD, MX, H);
    edge_expsum<<<nb(Etot * H, TB), TB, 0, stream>>>(ei, E, Etot, AS, AD, MX, ALPHA, DN, H);
    edge_aggregate<<<nb(Etot * H * (C / 4), TB), TB, 0, stream>>>(ei, E, Etot, ALPHA, DN, H1, AGG, H, C);
    bias_elu<<<nb(N * H * C, TB), TB, 0, stream>>>(AGG, b1, H1, N * H * C, H * C);  // H1 := elu(agg+b1)

    // ---------------- Layer 2 (H=1, C=64) ----------------
    float* H2    = AGG;                 // N*64 (AGG region now free)
    float* H2agg = AGG + (size_t)N * C; // N*64
    // grid: (625 M-groups, 1 N-strip of 64), whole B2 staged through LDS
    gemm_wmma_f32<4, H * C, C><<<dim3(N / 16 / 5, 1), 160, 0, stream>>>(H1, W2, H2);
    attn_coeff<<<nb(N, TB), TB, 0, stream>>>(H2, aS2, aD2, AS, AD, N, 1, C);
    fill_f32<<<nb(N, TB), TB, 0, stream>>>(MX, -INFINITY, N);
    fill_f32<<<nb(N, TB), TB, 0, stream>>>(DN, 0.f, N);
    fill_f32<<<nb(N * C, TB), TB, 0, stream>>>(H2agg, 0.f, N * C);
    edge_max   <<<nb(Etot, TB), TB, 0, stream>>>(ei, E, Etot, AS, AD, MX, 1);
    edge_expsum<<<nb(Etot, TB), TB, 0, stream>>>(ei, E, Etot, AS, AD, MX, ALPHA, DN, 1);
    edge_aggregate<<<nb(Etot * (C / 4), TB), TB, 0, stream>>>(ei, E, Etot, ALPHA, DN, H2, H2agg, 1, C);
    bias_elu<<<nb(N * C, TB), TB, 0, stream>>>(H2agg, b2, H2, N * C, C);            // H2 := elu(agg+b2)

    // ---------------- Pool + MLP head ----------------
    fill_f32<<<nb(B * C, TB), TB, 0, stream>>>(POOL, 0.f, B * C);
    pool_add<<<nb(N * C, TB), TB, 0, stream>>>(H2, batch, POOL, N, C);
    head_kernel<<<1, B, 0, stream>>>(POOL, l1w, l1b, l2w, l2b, out);
}